// Tensor_Fusion_7765300871202
// MI455X (gfx1250) — compile-verified
//
#include <hip/hip_runtime.h>

// ---------------------------------------------------------------------------
// Tensor fusion: o123 = o1 (x) o2 (x) o3 (rank-1, never materialized),
// out = relu(relu(o123 @ We1.T + be1) @ We2.T + be2).
// Dominant cost: streaming We1 (2.198 GB f32) once -> ~94us at 23.3 TB/s.
// Big GEMM runs on V_WMMA_F32_16X16X4_F32 with o123 built on the fly in LDS.
// ---------------------------------------------------------------------------

namespace {
constexpr int      kD        = 128;
constexpr int      kDP       = 129;
constexpr unsigned kFused    = 129u * 129u * 129u;   // 2,146,689
constexpr int      kNIJ      = 129 * 129;            // 16,641
constexpr int      kB        = 32;
constexpr int      kNOut     = 256;
constexpr int      kOPad     = 132;                  // o vectors padded [32][132]
constexpr int      kChunk    = 448;                  // p-values staged per chunk
constexpr int      kKSteps   = kChunk / 4;           // 112 wmma K-steps per tile
constexpr int      kLdsStr   = 452;                  // floats; 452 % 64 == 4 -> bank-conflict-free b64 reads
constexpr int      kNChunks  = (int)((kFused + kChunk - 1) / kChunk); // 4792
constexpr int      kMaxWG    = 1024;
}

typedef float v2f __attribute__((ext_vector_type(2)));
typedef float v8f __attribute__((ext_vector_type(8)));

__device__ __forceinline__ v8f wmma_f32_k4(v2f a, v2f b, v8f c) {
  // D(16x16,f32) = A(16x4,f32) x B(4x16,f32) + C
  return __builtin_amdgcn_wmma_f32_16x16x4_f32(false, a, false, b, (short)0, c,
                                               false, false);
}

// ---------------------------------------------------------------------------
// Kernel 1: one gating branch. 1 block x 128 threads; thread t owns column t.
// o_out layout: [32][132], col 128 = appended ones, cols 129..131 = 0 pad.
// ---------------------------------------------------------------------------
__global__ __launch_bounds__(128) void fusion_branch(
    const float* __restrict__ v1, const float* __restrict__ v2,
    const float* __restrict__ v3, const float* __restrict__ vb,
    const float* __restrict__ Wh, const float* __restrict__ bh,
    const float* __restrict__ Wz, const float* __restrict__ bz,
    const float* __restrict__ Wo, const float* __restrict__ bo,
    float* __restrict__ o_out) {
  __shared__ float sh[kB][kD];
  __shared__ float sg[kB][kD];
  const int t = threadIdx.x;  // 0..127
  const float* whr = Wh + t * kD;
  const float* wzr = Wz + t * 3 * kD;
  for (int b = 0; b < kB; ++b) {
    float ah = bh[t];
    float az = bz[t];
#pragma unroll 4
    for (int i = 0; i < kD; ++i) {
      ah += vb[b * kD + i] * whr[i];
      az += v1[b * kD + i] * wzr[i] + v2[b * kD + i] * wzr[kD + i] +
            v3[b * kD + i] * wzr[2 * kD + i];
    }
    sh[b][t] = fmaxf(ah, 0.0f);
    sg[b][t] = az;
  }
  __syncthreads();
  for (int b = 0; b < kB; ++b) {
    float z = sg[b][t];
    sg[b][t] = sh[b][t] * (1.0f / (1.0f + __expf(-z)));  // sigmoid(z)*h
  }
  __syncthreads();
  const float* wor = Wo + t * kD;
  for (int b = 0; b < kB; ++b) {
    float acc = bo[t];
#pragma unroll 4
    for (int i = 0; i < kD; ++i) acc += sg[b][i] * wor[i];
    o_out[b * kOPad + t] = fmaxf(acc, 0.0f);
  }
  if (t < 4) {
    for (int b = 0; b < kB; ++b)
      o_out[b * kOPad + kD + t] = (t == 0) ? 1.0f : 0.0f;  // ones col + zero pad
  }
}

// ---------------------------------------------------------------------------
// Kernel 2: o12[b][ij] = o1[b][i] * o2[b][j]   (2.1 MB, stays hot in L2)
// ---------------------------------------------------------------------------
__global__ __launch_bounds__(256) void o12_kernel(const float* __restrict__ o1,
                                                  const float* __restrict__ o2,
                                                  float* __restrict__ o12) {
  unsigned idx = blockIdx.x * 256u + threadIdx.x;
  if (idx >= (unsigned)(kB * kNIJ)) return;
  unsigned b = idx / kNIJ, ij = idx % kNIJ;
  unsigned i = ij / kDP, j = ij % kDP;
  o12[idx] = o1[b * kOPad + i] * o2[b * kOPad + j];
}

// ---------------------------------------------------------------------------
// Kernel 3: the big GEMM.  partial[wg][n][b] = sum_{p in wg's chunks}
//           We1[n,p] * o12[b,p/129] * o3[b,p%129]
// 8 waves/block: wave = (ntg 0..3, bhalf 0..1); each wave owns 4 WMMA tiles
// (64 n-rows x 16 batch cols) with chained f32 accumulation across all chunks.
// ---------------------------------------------------------------------------
__global__ __launch_bounds__(256) void big_gemm(const float* __restrict__ We1,
                                                const float* __restrict__ o12,
                                                const float* __restrict__ o3,
                                                float* __restrict__ partial,
                                                int nwg) {
  __shared__ float sb[kB * kLdsStr];  // 57,856 B staged o123 slice
  const int tid   = threadIdx.x;
  const int lane  = tid & 31;
  const int wave  = tid >> 5;
  const int ln    = lane & 15;
  const int khalf = (lane >> 4) << 1;   // lanes 0-15: K+0/K+1, lanes 16-31: K+2/K+3
  const int bhalf = wave & 1;
  const int ntg   = wave >> 1;          // 0..3 -> n-tiles [ntg*4, ntg*4+4)
  const int bidx  = bhalf * 16 + ln;    // this lane's batch column (B layout N=lane%16)

  v8f acc0 = {}, acc1 = {}, acc2 = {}, acc3 = {};

  const float* rowp[4];
#pragma unroll
  for (int t = 0; t < 4; ++t) {
    int n = (ntg * 4 + t) * 16 + ln;    // A layout: lane%16 = matrix row M
    rowp[t] = We1 + (size_t)n * kFused;
  }
  const float* sbp = sb + bidx * kLdsStr + khalf;

  for (int c = blockIdx.x; c < kNChunks; c += nwg) {
    const unsigned pbase = (unsigned)c * kChunk;
    __syncthreads();  // previous chunk's LDS reads done
    {   // stage o123[b][pl] = o12[b][p/129]*o3[b][p%129] (0 beyond FUSED)
      const int b0 = tid >> 3;
      const int pl0 = tid & 7;
      const float* o12b = o12 + b0 * kNIJ;
      const float* o3b  = o3 + b0 * kOPad;
      float* dst = sb + b0 * kLdsStr;
#pragma unroll 4
      for (int s = 0; s < kChunk / 8; ++s) {
        int pl = pl0 + 8 * s;
        unsigned p = pbase + (unsigned)pl;
        float v = 0.0f;
        if (p < kFused) {
          unsigned ij = p / 129u;
          unsigned k  = p - ij * 129u;
          v = o12b[ij] * o3b[k];
        }
        dst[pl] = v;
      }
    }
    __syncthreads();

    if (pbase + kChunk <= kFused) {     // fast path: all addresses in-bounds
      const float* a0 = rowp[0] + pbase + khalf;
      const float* a1 = rowp[1] + pbase + khalf;
      const float* a2 = rowp[2] + pbase + khalf;
      const float* a3 = rowp[3] + pbase + khalf;
#pragma unroll 4
      for (int ks = 0; ks < kKSteps; ++ks) {
        const int o = 4 * ks;
        v2f bf = *(const v2f*)(sbp + o);          // ds_load_b64, bank-conflict-free
        v2f A0; A0.x = a0[o]; A0.y = a0[o + 1];   // b32 pairs (row base may be 4B-aligned)
        v2f A1; A1.x = a1[o]; A1.y = a1[o + 1];
        v2f A2; A2.x = a2[o]; A2.y = a2[o + 1];
        v2f A3; A3.x = a3[o]; A3.y = a3[o + 1];
        acc0 = wmma_f32_k4(A0, bf, acc0);
        acc1 = wmma_f32_k4(A1, bf, acc1);
        acc2 = wmma_f32_k4(A2, bf, acc2);
        acc3 = wmma_f32_k4(A3, bf, acc3);
      }
    } else {                            // tail chunk: clamp, staged zeros kill garbage
      for (int ks = 0; ks < kKSteps; ++ks) {
        const int o = 4 * ks;
        v2f bf = *(const v2f*)(sbp + o);
        unsigned p0 = pbase + (unsigned)o + (unsigned)khalf;
        unsigned px = (p0 < kFused - 1u) ? p0 : (kFused - 1u);
        unsigned py = (p0 + 1u < kFused - 1u) ? (p0 + 1u) : (kFused - 1u);
        v2f A0; A0.x = rowp[0][px]; A0.y = rowp[0][py];
        v2f A1; A1.x = rowp[1][px]; A1.y = rowp[1][py];
        v2f A2; A2.x = rowp[2][px]; A2.y = rowp[2][py];
        v2f A3; A3.x = rowp[3][px]; A3.y = rowp[3][py];
        acc0 = wmma_f32_k4(A0, bf, acc0);
        acc1 = wmma_f32_k4(A1, bf, acc1);
        acc2 = wmma_f32_k4(A2, bf, acc2);
        acc3 = wmma_f32_k4(A3, bf, acc3);
      }
    }
  }

  // Write this workgroup's partial [256 n][32 b].
  // D layout: (VGPR r, lane l) -> M = r + 8*(l>=16), N = l%16.
  float* pb = partial + (size_t)blockIdx.x * (kNOut * kB);
  const int mofs = (lane >> 4) * 8;
  v8f* accs[4] = {&acc0, &acc1, &acc2, &acc3};
#pragma unroll
  for (int t = 0; t < 4; ++t) {
    int nbase = (ntg * 4 + t) * 16 + mofs;
    v8f c = *accs[t];
#pragma unroll
    for (int r = 0; r < 8; ++r) pb[(nbase + r) * kB + bidx] = c[r];
  }
}

// ---------------------------------------------------------------------------
// Kernel 4: deterministic reduction over partials + bias + relu.
// hidden[e] with e = n*32 + b. Fully coalesced (each wg slab contiguous).
// ---------------------------------------------------------------------------
__global__ __launch_bounds__(256) void reduce_bias_relu(
    const float* __restrict__ partial, const float* __restrict__ be1,
    float* __restrict__ hidden, int nwg) {
  int e = blockIdx.x * 256 + threadIdx.x;  // 0..8191
  int n = e >> 5;
  float s = be1[n];
  for (int w = 0; w < nwg; ++w) s += partial[(size_t)w * (kNOut * kB) + e];
  hidden[e] = fmaxf(s, 0.0f);
}

// ---------------------------------------------------------------------------
// Kernel 5: second layer, out[b][m] = relu(be2[m] + sum_n hidden[n][b]*We2[m][n])
// Grid: 32 blocks x 256 threads -> (8 m) x (32 b) per block, 256 m total.
// ---------------------------------------------------------------------------
__global__ __launch_bounds__(256) void mlp2(const float* __restrict__ hidden,
                                            const float* __restrict__ We2,
                                            const float* __restrict__ be2,
                                            float* __restrict__ out) {
  int b = threadIdx.x & 31;
  int m = blockIdx.x * 8 + (threadIdx.x >> 5);
  float acc = be2[m];
  const float* wr = We2 + m * kNOut;
#pragma unroll 4
  for (int n = 0; n < kNOut; ++n) acc += hidden[n * kB + b] * wr[n];
  out[b * kNOut + m] = fmaxf(acc, 0.0f);
}

// ---------------------------------------------------------------------------
extern "C" void kernel_launch(void* const* d_in, const int* in_sizes, int n_in,
                              void* d_out, int out_size, void* d_ws,
                              size_t ws_size, hipStream_t stream) {
  const float* vec1 = (const float*)d_in[0];
  const float* vec2 = (const float*)d_in[1];
  const float* vec3 = (const float*)d_in[2];
  const float* Wb[3][6];
  for (int r = 0; r < 3; ++r)
    for (int q = 0; q < 6; ++q) Wb[r][q] = (const float*)d_in[3 + r * 6 + q];
  const float* We1 = (const float*)d_in[21];
  const float* be1 = (const float*)d_in[22];
  const float* We2 = (const float*)d_in[23];
  const float* be2 = (const float*)d_in[24];

  float* ws = (float*)d_ws;
  float* o1 = ws;
  float* o2 = o1 + kB * kOPad;
  float* o3 = o2 + kB * kOPad;
  float* o12 = o3 + kB * kOPad;
  float* hidden = o12 + kB * kNIJ;
  size_t fixedf = (size_t)(3 * kB * kOPad + kB * kNIJ + kNOut * kB);
  fixedf = (fixedf + 63) & ~(size_t)63;
  float* partial = ws + fixedf;

  size_t availf = (ws_size / 4 > fixedf) ? (ws_size / 4 - fixedf) : 0;
  int nwg = (int)(availf / (size_t)(kNOut * kB));
  if (nwg > kMaxWG) nwg = kMaxWG;
  if (nwg < 1) nwg = 1;

  const float* vecs[3] = {vec1, vec2, vec3};
  float* os[3] = {o1, o2, o3};
  for (int r = 0; r < 3; ++r) {
    fusion_branch<<<1, 128, 0, stream>>>(vec1, vec2, vec3, vecs[r], Wb[r][0],
                                         Wb[r][1], Wb[r][2], Wb[r][3], Wb[r][4],
                                         Wb[r][5], os[r]);
  }
  o12_kernel<<<(kB * kNIJ + 255) / 256, 256, 0, stream>>>(o1, o2, o12);
  big_gemm<<<nwg, 256, 0, stream>>>(We1, o12, o3, partial, nwg);
  reduce_bias_relu<<<(kNOut * kB) / 256, 256, 0, stream>>>(partial, be1, hidden,
                                                           nwg);
  mlp2<<<32, 256, 0, stream>>>(hidden, We2, be2, (float*)d_out);
}